// DigitCaps_52132313039386
// MI455X (gfx1250) — compile-verified
//
#include <hip/hip_runtime.h>

// ---------------- problem constants (match reference) ----------------
#define BATCH   128
#define NCAPS   1152      // 6*6*32 input capsules
#define DIN     8
#define DNUM    16        // vector depth
#define CNUM    10        // output capsules
#define MT      10        // transforms per capsule
#define ITERS   3
#define JPW     16        // capsules processed per workgroup (amortize atomics)

// workspace layout (floats)
#define WS_DC    0        // [10][16] codebook
#define WS_MASS  160      // [10]
#define WS_WU    176      // [10][16]
#define WS_TOT   336

typedef __attribute__((ext_vector_type(2))) float v2f;
typedef __attribute__((ext_vector_type(8))) float v8f;

// ---------------------------------------------------------------------
// init: copy codebook into ws, zero accumulators, zero output sums
__global__ void caps_init(const float* __restrict__ dc_in,
                          float* __restrict__ ws,
                          float* __restrict__ out) {
    int i = blockIdx.x * blockDim.x + threadIdx.x;
    if (i < 160)                 ws[WS_DC + i] = dc_in[i];
    if (i >= 160 && i < WS_TOT)  ws[i] = 0.0f;               // mass + wu (+pad)
    if (i < BATCH * CNUM)        out[i] = 0.0f;              // sims sums
}

// ---------------------------------------------------------------------
// One full pass over all votes. MODE 0: routing iteration (accumulate
// mass / weighted_u). MODE 1: final pass (accumulate per-batch sims sums).
// Block = 320 threads = 10 waves; wave t owns transform t. Grid = 8 b-tiles
// x (1152/JPW) capsule groups = 576 workgroups.
template <int MODE>
__global__ __launch_bounds__(320)
void caps_pass(const float* __restrict__ x,   // [B,1152,8]
               const float* __restrict__ W,   // [1152,8,160]
               float* __restrict__ ws,
               float* __restrict__ out) {
    __shared__ float lds_dc[CNUM][DNUM];        // codebook
    __shared__ float lds_u[MT][16][DNUM];       // vote tiles (10 KB)
    __shared__ float lds_mass[CNUM];
    __shared__ float lds_wu[CNUM][DNUM];
    __shared__ float lds_osum[16][CNUM];

    const int tid = threadIdx.x;
    if (tid < 160) {
        lds_dc[tid >> 4][tid & 15] = ws[WS_DC + tid];
        lds_wu[tid >> 4][tid & 15] = 0.0f;
        lds_osum[tid / CNUM][tid % CNUM] = 0.0f;
    }
    if (tid < CNUM) lds_mass[tid] = 0.0f;
    __syncthreads();

    const int t    = tid >> 5;        // wave id -> transform index
    const int lane = tid & 31;
    const int m    = lane & 15;       // row (A) / col (B,D) within tile
    const int hi   = lane >> 4;       // K-half selector
    const int bt   = blockIdx.x & 7;  // b tile
    const int jg   = blockIdx.x >> 3; // capsule group
    const int b0   = bt * 16;

    for (int jj = 0; jj < JPW; ++jj) {
        const int j = jg * JPW + jj;

        // ---- A tile: x2[b0+m, j, k], K pair = {2*hi, 2*hi+1} (+4 for step 2)
        const float* xp = x + (size_t)(b0 + m) * (NCAPS * DIN) + (size_t)j * DIN;
        const v2f* xv = (const v2f*)(xp + 2 * hi);
        v2f a0 = xv[0];
        v2f a1 = xv[2];

        // ---- B tile: W[j, k, t*16 + m], rows strided by 160 floats
        const float* wp = W + (size_t)j * (DIN * MT * DNUM) + t * DNUM + m;
        v2f bm0, bm1;
        bm0.x = wp[(0 + 2 * hi) * (MT * DNUM)];
        bm0.y = wp[(1 + 2 * hi) * (MT * DNUM)];
        bm1.x = wp[(4 + 2 * hi) * (MT * DNUM)];
        bm1.y = wp[(5 + 2 * hi) * (MT * DNUM)];

        // ---- u tile = A(16x8) * B(8x16) via two fp32 WMMAs (K=4 each)
        v8f acc = {};
        acc = __builtin_amdgcn_wmma_f32_16x16x4_f32(false, a0, false, bm0,
                                                    (short)0, acc, false, false);
        acc = __builtin_amdgcn_wmma_f32_16x16x4_f32(false, a1, false, bm1,
                                                    (short)0, acc, false, false);

        // D layout: VGPR r, lanes 0-15 -> row r, lanes 16-31 -> row r+8; col = m
#pragma unroll
        for (int r = 0; r < 8; ++r)
            lds_u[t][r + 8 * hi][m] = acc[r];
        __syncthreads();

        // ---- per-vote sims / argmax (160 votes per tile set)
        if (tid < 16 * MT) {
            const int bl = tid / MT;   // local batch row
            const int tt = tid % MT;   // transform
            float u[DNUM];
#pragma unroll
            for (int d = 0; d < DNUM; ++d) u[d] = lds_u[tt][bl][d];

            float best = -3.402823466e38f;
            int   bi   = 0;
            float sims[CNUM];
#pragma unroll
            for (int c = 0; c < CNUM; ++c) {
                float s = 0.0f;
#pragma unroll
                for (int d = 0; d < DNUM; ++d) s += u[d] * lds_dc[c][d];
                sims[c] = s;
                if (s > best) { best = s; bi = c; }   // first-max tie-break
            }
            if (MODE == 0) {
                atomicAdd(&lds_mass[bi], 1.0f);
#pragma unroll
                for (int d = 0; d < DNUM; ++d) atomicAdd(&lds_wu[bi][d], u[d]);
            } else {
#pragma unroll
                for (int c = 0; c < CNUM; ++c) atomicAdd(&lds_osum[bl][c], sims[c]);
            }
        }
        __syncthreads();   // protect lds_u before next capsule overwrites it
    }

    // ---- single flush of block-local accumulators to global
    if (MODE == 0) {
        if (tid < CNUM) atomicAdd(&ws[WS_MASS + tid], lds_mass[tid]);
        if (tid < 160)  atomicAdd(&ws[WS_WU + tid], lds_wu[tid >> 4][tid & 15]);
    } else {
        if (tid < 160) {
            const int bl = tid / CNUM, c = tid % CNUM;
            atomicAdd(&out[(b0 + bl) * CNUM + c], lds_osum[bl][c]);
        }
    }
}

// ---------------------------------------------------------------------
// codebook update: dc += (wu - dc*mass) * (LR*THETA/(B*M)); reset accumulators
__global__ void caps_update(float* __restrict__ ws) {
    const int tid = threadIdx.x;           // 160 threads
    const float scale = 1.0f / (float)(BATCH * NCAPS * MT);
    float nd = 0.0f;
    if (tid < 160) {
        const float mass = ws[WS_MASS + (tid >> 4)];
        const float dc   = ws[WS_DC + tid];
        const float wu   = ws[WS_WU + tid];
        nd = dc + (wu - dc * mass) * scale;
    }
    __syncthreads();                       // all reads done before writes
    if (tid < 160) {
        ws[WS_DC + tid] = nd;
        ws[WS_WU + tid] = 0.0f;
    }
    if (tid < CNUM) ws[WS_MASS + tid] = 0.0f;
}

// ---------------------------------------------------------------------
// finalize: out[:1280] = sims_sum / M ; out[1280:1440] = final codebook
__global__ void caps_finalize(const float* __restrict__ ws,
                              float* __restrict__ out) {
    const int i = blockIdx.x * blockDim.x + threadIdx.x;
    const float invM = 1.0f / (float)(NCAPS * MT);
    if (i < BATCH * CNUM)                        out[i] *= invM;
    else if (i < BATCH * CNUM + 160)             out[i] = ws[WS_DC + (i - BATCH * CNUM)];
}

// ---------------------------------------------------------------------
extern "C" void kernel_launch(void* const* d_in, const int* in_sizes, int n_in,
                              void* d_out, int out_size, void* d_ws, size_t ws_size,
                              hipStream_t stream) {
    (void)in_sizes; (void)n_in; (void)out_size; (void)ws_size;
    const float* x  = (const float*)d_in[0];   // [128,6,6,32,8]
    const float* W  = (const float*)d_in[1];   // [1152,8,160]
    const float* dc = (const float*)d_in[2];   // [10,16]
    float* out = (float*)d_out;                // [1280 + 160]
    float* ws  = (float*)d_ws;

    caps_init<<<6, 256, 0, stream>>>(dc, ws, out);

    const dim3 grid(8 * (NCAPS / JPW));        // 576 workgroups
    const dim3 block(32 * MT);                 // 10 waves of 32
    for (int it = 0; it < ITERS; ++it) {
        caps_pass<0><<<grid, block, 0, stream>>>(x, W, ws, out);
        caps_update<<<1, 160, 0, stream>>>(ws);
    }
    caps_pass<1><<<grid, block, 0, stream>>>(x, W, ws, out);
    caps_finalize<<<6, 256, 0, stream>>>(ws, out);
}